// SparseConvNet_35562329211049
// MI455X (gfx1250) — compile-verified
//
#include <hip/hip_runtime.h>

typedef __attribute__((ext_vector_type(2))) float v2f;
typedef __attribute__((ext_vector_type(8))) float v8f;

#define DGRID 64
#define NCELL (64 * 64 * 64)

// ---------------- grid build ----------------
__global__ void fill_i32(int* __restrict__ p, int n, int v) {
  int i = blockIdx.x * blockDim.x + threadIdx.x;
  if (i < n) p[i] = v;
}

__global__ void scatter_grid(const int* __restrict__ coords, int n, int* __restrict__ grid) {
  int i = blockIdx.x * blockDim.x + threadIdx.x;
  if (i < n) {
    int z = coords[i * 4 + 1], y = coords[i * 4 + 2], x = coords[i * 4 + 3];
    grid[(z * DGRID + y) * DGRID + x] = i;
  }
}

// ---------------- sparse gather-GEMM via f32 WMMA ----------------
// Wave handles one 16-row output tile x 64 output channels.
// A(16x4) fp32 fragment: lane<16 -> K=0,1 ; lane>=16 -> K=2,3 (per lane's row M=lane&15).
// B(4x16) fragment from LDS weight slab (layout [K][Cout] row-major).
template <int CIN>
__global__ __launch_bounds__(256) void sparse_conv_wmma(
    const float* __restrict__ feats, const int* __restrict__ grid,
    const int* __restrict__ ocoords, const float* __restrict__ W,
    const float* __restrict__ bias, float* __restrict__ out, int nOut) {
  __shared__ float wlds[CIN * 64];
  const int lane = threadIdx.x & 31;
  const int wave = threadIdx.x >> 5;
  const int m = lane & 15;
  const int koff = (lane < 16) ? 0 : 2;
  const int tile = blockIdx.x * 8 + wave;

  int orow = tile * 16 + m;
  bool rowOk = orow < nOut;
  int oz = 0, oy = 0, ox = 0;
  if (rowOk) {
    oz = ocoords[orow * 4 + 1];
    oy = ocoords[orow * 4 + 2];
    ox = ocoords[orow * 4 + 3];
  }

  v8f acc[4];
  for (int t = 0; t < 4; ++t)
    for (int v = 0; v < 8; ++v) acc[t][v] = 0.0f;

  for (int k = 0; k < 27; ++k) {
    // stage this offset's (CIN x 64) weight slab in LDS
    const float* wk = W + (size_t)k * CIN * 64;
    for (int i = threadIdx.x; i < CIN * 64; i += 256) wlds[i] = wk[i];
    __syncthreads();

    int dz = k / 9 - 1, dy = (k / 3) % 3 - 1, dx = k % 3 - 1;
    int nz = oz + dz, ny = oy + dy, nx = ox + dx;
    int row = -1;
    if (rowOk && nz >= 0 && nz < DGRID && ny >= 0 && ny < DGRID && nx >= 0 && nx < DGRID)
      row = grid[(nz * DGRID + ny) * DGRID + nx];
    const float* arow = feats + (size_t)(row < 0 ? 0 : row) * CIN;
    bool have = (row >= 0);

#pragma unroll 4
    for (int kc = 0; kc < CIN / 4; ++kc) {
      v2f a;
      a[0] = have ? arow[kc * 4 + koff] : 0.0f;
      a[1] = have ? arow[kc * 4 + koff + 1] : 0.0f;
      const float* b0 = &wlds[(kc * 4 + koff) * 64 + m];
#pragma unroll
      for (int t = 0; t < 4; ++t) {
        v2f b;
        b[0] = b0[t * 16];
        b[1] = b0[64 + t * 16];
        acc[t] = __builtin_amdgcn_wmma_f32_16x16x4_f32(
            false, a, false, b, (short)0, acc[t], false, false);
      }
    }
    __syncthreads();
  }

  // D layout: VGPR v -> M = v (lanes 0-15) / v+8 (lanes 16-31); N = lane&15
  for (int t = 0; t < 4; ++t) {
    int col = t * 16 + m;
    float bv = bias[col];
#pragma unroll
    for (int v = 0; v < 8; ++v) {
      int r = tile * 16 + ((lane < 16) ? v : v + 8);
      if (r < nOut) out[(size_t)r * 64 + col] = acc[t][v] + bv;
    }
  }
}

// ---------------- flat-chunk groupnorm + relu (matches the reshape trick) ----------------
__global__ __launch_bounds__(256) void group_bn_relu(
    const float* __restrict__ f, float* __restrict__ o,
    const float* __restrict__ gamma, const float* __restrict__ beta, int gs) {
  __shared__ float ssum[256], ssq[256];
  int g = blockIdx.x;
  const float* p = f + (size_t)g * gs;
  float s = 0.f, q = 0.f;
  for (int i = threadIdx.x; i < gs; i += 256) {
    float v = p[i];
    s += v;
    q += v * v;
  }
  ssum[threadIdx.x] = s;
  ssq[threadIdx.x] = q;
  __syncthreads();
  for (int off = 128; off > 0; off >>= 1) {
    if (threadIdx.x < off) {
      ssum[threadIdx.x] += ssum[threadIdx.x + off];
      ssq[threadIdx.x] += ssq[threadIdx.x + off];
    }
    __syncthreads();
  }
  float inv = 1.0f / (float)gs;
  float mean = ssum[0] * inv;
  float var = ssq[0] * inv - mean * mean;
  float rstd = rsqrtf(var + 1e-5f);
  float ga = gamma[g], be = beta[g];
  float* po = o + (size_t)g * gs;
  for (int i = threadIdx.x; i < gs; i += 256) {
    float v = (p[i] - mean) * rstd * ga + be;
    po[i] = v > 0.f ? v : 0.f;
  }
}

// ---------------- pack wf[:, :, 1, 1, :] -> wk[k][o*64+i] ----------------
__global__ void pack_wf(const float* __restrict__ wf, float* __restrict__ wk) {
  int i = blockIdx.x * blockDim.x + threadIdx.x;
  if (i < 3 * 64 * 64) {
    int k = i / 4096;
    int rem = i - k * 4096;
    int o = rem >> 6, ci = rem & 63;
    wk[i] = wf[o * 1728 + ci * 27 + 12 + k];  // [o][i][1][1][k]
  }
}

// ---------------- 3-tap conv along N2 as 3 shifted 64x64 GEMMs (WMMA) ----------------
__global__ __launch_bounds__(256) void final_conv_wmma(
    const float* __restrict__ y, const float* __restrict__ wk,
    const float* __restrict__ bf, float* __restrict__ z, int n2) {
  const int lane = threadIdx.x & 31;
  const int wave = threadIdx.x >> 5;
  const int m = lane & 15;
  const int koff = (lane < 16) ? 0 : 2;
  int wtile = blockIdx.x * 8 + wave;
  int w0 = wtile * 16;
  if (w0 >= n2) return;  // wave-uniform
  int wcol = w0 + m;

  v8f acc[4];
  for (int t = 0; t < 4; ++t)
    for (int v = 0; v < 8; ++v) acc[t][v] = 0.0f;

  for (int k = 0; k < 3; ++k) {
    const float* wkk = wk + k * 4096;
    int ws = wcol + k - 1;
    bool inb = (ws >= 0) && (ws < n2);
    size_t wsz = (size_t)(inb ? ws : 0);
#pragma unroll 4
    for (int kc = 0; kc < 16; ++kc) {
      int r0 = kc * 4 + koff;
      v2f b;
      b[0] = inb ? y[(size_t)r0 * n2 + wsz] : 0.0f;
      b[1] = inb ? y[(size_t)(r0 + 1) * n2 + wsz] : 0.0f;
#pragma unroll
      for (int t = 0; t < 4; ++t) {
        v2f a;
        a[0] = wkk[(t * 16 + m) * 64 + r0];
        a[1] = wkk[(t * 16 + m) * 64 + r0 + 1];
        acc[t] = __builtin_amdgcn_wmma_f32_16x16x4_f32(
            false, a, false, b, (short)0, acc[t], false, false);
      }
    }
  }

  for (int t = 0; t < 4; ++t) {
#pragma unroll
    for (int v = 0; v < 8; ++v) {
      int o = t * 16 + ((lane < 16) ? v : v + 8);
      int wc = w0 + m;
      if (wc < n2) z[(size_t)o * n2 + wc] = acc[t][v] + bf[o];
    }
  }
}

// ---------------- bias-included z -> linear interp along N2, broadcast (30,30) ----------------
__global__ void interp_out(const float* __restrict__ z, float* __restrict__ out, int n2) {
  int i = blockIdx.x * blockDim.x + threadIdx.x;
  if (i >= 64 * 30 * 30 * 30) return;
  int mw = i % 30;
  int c = i / 27000;
  float src = (mw + 0.5f) * ((float)n2 / 30.0f) - 0.5f;
  src = fminf(fmaxf(src, 0.0f), (float)(n2 - 1));
  int lo = (int)floorf(src);
  int hi = min(lo + 1, n2 - 1);
  float t = src - (float)lo;
  out[i] = z[(size_t)c * n2 + lo] * (1.0f - t) + z[(size_t)c * n2 + hi] * t;
}

extern "C" void kernel_launch(void* const* d_in, const int* in_sizes, int n_in,
                              void* d_out, int out_size, void* d_ws, size_t ws_size,
                              hipStream_t stream) {
  const float* feats = (const float*)d_in[0];
  const float* w1 = (const float*)d_in[1];
  const float* b1 = (const float*)d_in[2];
  const float* g1 = (const float*)d_in[3];
  const float* be1 = (const float*)d_in[4];
  const float* w2 = (const float*)d_in[5];
  const float* b2 = (const float*)d_in[6];
  const float* g2 = (const float*)d_in[7];
  const float* be2 = (const float*)d_in[8];
  const float* wf = (const float*)d_in[9];
  const float* bf = (const float*)d_in[10];
  const int* c0 = (const int*)d_in[11];
  const int* c1 = (const int*)d_in[12];
  const int* c2 = (const int*)d_in[13];

  int n0 = in_sizes[0] / 128;
  int n1 = in_sizes[12] / 4;
  int n2 = in_sizes[13] / 4;

  char* ws = (char*)d_ws;
  int* grid0 = (int*)ws;            ws += (size_t)NCELL * 4;
  int* grid1 = (int*)ws;            ws += (size_t)NCELL * 4;
  float* wkp = (float*)ws;          ws += 3 * 64 * 64 * 4;
  size_t bigN = (size_t)((n2 > n1 ? n2 : n1)) * 64;
  bigN = (bigN + 63) & ~(size_t)63;
  float* bufA = (float*)ws;         ws += bigN * 4;   // f1 -> f2 -> z
  float* bufB = (float*)ws;                            // x1 -> y

  // hash grids
  fill_i32<<<(NCELL + 255) / 256, 256, 0, stream>>>(grid0, NCELL, -1);
  fill_i32<<<(NCELL + 255) / 256, 256, 0, stream>>>(grid1, NCELL, -1);
  scatter_grid<<<(n0 + 255) / 256, 256, 0, stream>>>(c0, n0, grid0);
  scatter_grid<<<(n1 + 255) / 256, 256, 0, stream>>>(c1, n1, grid1);

  // layer 1: sparse conv (128 -> 64) + flat groupnorm/relu
  sparse_conv_wmma<128><<<(n1 + 127) / 128, 256, 0, stream>>>(
      feats, grid0, c1, w1, b1, bufA, n1);
  group_bn_relu<<<64, 256, 0, stream>>>(bufA, bufB, g1, be1, n1);

  // layer 2: sparse conv (64 -> 64) + flat groupnorm/relu
  sparse_conv_wmma<64><<<(n2 + 127) / 128, 256, 0, stream>>>(
      bufB, grid1, c2, w2, b2, bufA, n2);
  group_bn_relu<<<64, 256, 0, stream>>>(bufA, bufB, g2, be2, n2);

  // center-slice 3-tap conv along N2 (bias folded in), then interp + broadcast
  pack_wf<<<(3 * 64 * 64 + 255) / 256, 256, 0, stream>>>(wf, wkp);
  final_conv_wmma<<<(n2 + 127) / 128, 256, 0, stream>>>(bufB, wkp, bf, bufA, n2);

  int totOut = 64 * 30 * 30 * 30;
  interp_out<<<(totOut + 255) / 256, 256, 0, stream>>>(bufA, (float*)d_out, n2);
}